// FindInstancePeaks_21148418966129
// MI455X (gfx1250) — compile-verified
//
#include <hip/hip_runtime.h>
#include <hip/hip_bf16.h>

typedef __attribute__((ext_vector_type(2))) float v2f;
typedef __attribute__((ext_vector_type(8))) float v8f;

// ---- order-preserving float <-> u32 mapping for atomic max keys ----
__device__ __forceinline__ unsigned ford(float f) {
  unsigned u = __float_as_uint(f);
  return (u & 0x80000000u) ? ~u : (u | 0x80000000u);
}
__device__ __forceinline__ float finv(unsigned u) {
  unsigned v = (u & 0x80000000u) ? (u & 0x7fffffffu) : ~u;
  return __uint_as_float(v);
}

#define NKEYS (128 * 17)

// -------------------------------------------------------------------
// Prep: zero argmax keys; build W2T[n][k] (n=0..31 padded, k=tap*32+ci)
// -------------------------------------------------------------------
__global__ void prep_kernel(const float* __restrict__ W2,
                            unsigned long long* __restrict__ keys,
                            float* __restrict__ W2T) {
  int i = blockIdx.x * blockDim.x + threadIdx.x;
  if (i < NKEYS) keys[i] = 0ULL;
  if (i < 32 * 288) {
    int n = i / 288, k = i % 288;
    int tap = k >> 5, c = k & 31;          // tap = ky*3+kx
    W2T[i] = (n < 17) ? W2[(tap * 32 + c) * 17 + n] : 0.0f;
  }
}

// -------------------------------------------------------------------
// Fused conv1+relu -> conv2 (WMMA f32) -> per-channel argmax (atomic)
// Block = 128 threads (4 waves); tile = 16x16 output pixels of one image.
// -------------------------------------------------------------------
__global__ __launch_bounds__(128) void convpeak_kernel(
    const float* __restrict__ crops, const float* __restrict__ W1,
    const float* __restrict__ b1, const float* __restrict__ b2,
    const float* __restrict__ W2T, unsigned long long* __restrict__ keys) {
  __shared__ __align__(16) float cropT[37 * 38];      // crop tile (zero-padded)
  __shared__ __align__(16) float hT[18 * 18 * 34];    // h tile, 34-dword pixel stride
  __shared__ __align__(16) float b2s[32];

  const int tid = threadIdx.x;
  const int blk = blockIdx.x;
  const int b = blk / 36;
  const int t = blk % 36;
  const int ty = t / 6, tx = t % 6;
  const int r0 = ty * 16, c0 = tx * 16;

  // ---- stage crop tile: rows 2*r0-2 .. +36, cols 2*c0-2 .. +36 ----
  {
    const int cy0 = 2 * r0 - 2, cx0 = 2 * c0 - 2;
    for (int i = tid; i < 37 * 37; i += 128) {
      int cy = i / 37, cx = i % 37;
      int gy = cy0 + cy, gx = cx0 + cx;
      float v = 0.0f;
      if (gy >= 0 && gy < 192 && gx >= 0 && gx < 192)
        v = crops[(b * 192 + gy) * 192 + gx];
      cropT[cy * 38 + cx] = v;
    }
    if (tid < 32) b2s[tid] = (tid < 17) ? b2[tid] : 0.0f;
  }

  // per-thread conv1 weights for its fixed channel ci = tid&31
  const int ci = tid & 31;
  float w1r[9];
#pragma unroll
  for (int q = 0; q < 9; ++q) w1r[q] = W1[q * 32 + ci];
  const float b1r = b1[ci];
  __syncthreads();

  // ---- conv1 + relu into hT (18x18 pixels, 32 ch); zero outside image ----
  // each wave computes one pixel per iteration (lanes = channels, broadcast reads)
  for (int p = (tid >> 5); p < 18 * 18; p += 4) {
    const int hyL = p / 18, hxL = p % 18;
    const int hy = r0 - 1 + hyL, hx = c0 - 1 + hxL;
    float h = 0.0f;
    if (hy >= 0 && hy < 96 && hx >= 0 && hx < 96) {
      float s = b1r;
#pragma unroll
      for (int qy = 0; qy < 3; ++qy)
#pragma unroll
        for (int qx = 0; qx < 3; ++qx)
          s = fmaf(cropT[(2 * hyL + qy) * 38 + (2 * hxL + qx)], w1r[qy * 3 + qx], s);
      h = fmaxf(s, 0.0f);
    }
    hT[p * 34 + ci] = h;
  }
  __syncthreads();

  // ---- conv2 as implicit GEMM via v_wmma_f32_16x16x4_f32 ----
  const int w = tid >> 5;      // wave id (0..3) -> 4 rows each
  const int lane = tid & 31;
  const int m = lane & 15;     // M (pixel) for A, N (channel) for B/C
  const int half = lane >> 4;  // K sub-offset selector

  unsigned long long bkey = 0ULL, bkey16 = 0ULL;

  for (int rr = 0; rr < 4; ++rr) {
    const int row = w * 4 + rr;
    v8f acc0, acc1;
    const float i0 = b2s[m], i1 = b2s[16 + m];
#pragma unroll
    for (int j = 0; j < 8; ++j) { acc0[j] = i0; acc1[j] = i1; }

#pragma unroll
    for (int tap = 0; tap < 9; ++tap) {
      const int dy = tap / 3, dx = tap % 3;
      const float* aBase = &hT[((row + dy) * 18 + (m + dx)) * 34 + 2 * half];
      const float* f0Base = &W2T[m * 288 + tap * 32 + 2 * half];
      const float* f1Base = &W2T[(16 + m) * 288 + tap * 32 + 2 * half];
#pragma unroll
      for (int g = 0; g < 8; ++g) {
        v2f a = *(const v2f*)(aBase + 4 * g);
        v2f f0 = *(const v2f*)(f0Base + 4 * g);
        v2f f1 = *(const v2f*)(f1Base + 4 * g);
        acc0 = __builtin_amdgcn_wmma_f32_16x16x4_f32(false, a, false, f0,
                                                     (short)0, acc0, false, false);
        acc1 = __builtin_amdgcn_wmma_f32_16x16x4_f32(false, a, false, f1,
                                                     (short)0, acc1, false, false);
      }
    }

    // ---- fold this row's 16 pixels into running per-channel argmax ----
    const int ybase = (r0 + row) * 96;
#pragma unroll
    for (int j = 0; j < 8; ++j) {
      const unsigned idx = (unsigned)(ybase + c0 + j + 8 * half);
      const unsigned long long k0 =
          ((unsigned long long)ford(acc0[j]) << 32) | (unsigned)(~idx);
      if (k0 > bkey) bkey = k0;
      const unsigned long long k1 =
          ((unsigned long long)ford(acc1[j]) << 32) | (unsigned)(~idx);
      if (k1 > bkey16) bkey16 = k1;
    }
  }

  // merge half-waves (lane L with L^16 share the same channel)
  unsigned long long o = __shfl_xor(bkey, 16, 32);
  if (o > bkey) bkey = o;
  o = __shfl_xor(bkey16, 16, 32);
  if (o > bkey16) bkey16 = o;

  if (lane < 16) atomicMax(&keys[b * 17 + lane], bkey);
  if (lane == 0) atomicMax(&keys[b * 17 + 16], bkey16);
}

// -------------------------------------------------------------------
// Refinement: recompute the 4 neighbor confmap values per peak,
// quarter-pixel offset, threshold, scale by OUTPUT_STRIDE=2, pack out.
// One 32-lane block per (b, n); lane = input channel ci.
// -------------------------------------------------------------------
__global__ __launch_bounds__(32) void refine_kernel(
    const float* __restrict__ crops, const float* __restrict__ W1,
    const float* __restrict__ b1, const float* __restrict__ W2,
    const float* __restrict__ b2, const unsigned long long* __restrict__ keys,
    float* __restrict__ out) {
  const int bid = blockIdx.x;
  const int b = bid / 17, n = bid % 17;
  const int lane = threadIdx.x;  // ci

  const unsigned long long key = keys[bid];
  const float val = finv((unsigned)(key >> 32));
  const unsigned idx = ~(unsigned)(key & 0xffffffffu);
  const int yi = (int)(idx / 96), xi = (int)(idx % 96);

  const int dyv[4] = {0, 0, 1, -1};
  const int dxv[4] = {1, -1, 0, 0};
  float nb[4];

  for (int tt = 0; tt < 4; ++tt) {
    int y = min(95, max(0, yi + dyv[tt]));
    int x = min(95, max(0, xi + dxv[tt]));
    float part = 0.0f;
#pragma unroll
    for (int ky = 0; ky < 3; ++ky)
#pragma unroll
      for (int kx = 0; kx < 3; ++kx) {
        const int hy = y + ky - 1, hx = x + kx - 1;  // conv2 pad_lo = 1
        float h = 0.0f;
        if (hy >= 0 && hy < 96 && hx >= 0 && hx < 96) {
          float s = b1[lane];
#pragma unroll
          for (int qy = 0; qy < 3; ++qy)
#pragma unroll
            for (int qx = 0; qx < 3; ++qx) {
              const int cy = 2 * hy + qy, cx = 2 * hx + qx;  // conv1 pad_lo = 0
              if (cy < 192 && cx < 192)
                s = fmaf(crops[(b * 192 + cy) * 192 + cx],
                         W1[(qy * 3 + qx) * 32 + lane], s);
            }
          h = fmaxf(s, 0.0f);
        }
        part = fmaf(h, W2[((ky * 3 + kx) * 32 + lane) * 17 + n], part);
      }
    for (int off = 16; off > 0; off >>= 1) part += __shfl_down(part, off, 32);
    nb[tt] = part;  // valid on lane 0
  }

  if (lane == 0) {
    const float bias = b2[n];
    const float right = nb[0] + bias, left = nb[1] + bias;
    const float down = nb[2] + bias, up = nb[3] + bias;
    const float dxo = 0.25f * (float)((right > left) - (right < left));
    const float dyo = 0.25f * (float)((down > up) - (down < up));
    float px = ((float)xi + dxo) * 2.0f;
    float py = ((float)yi + dyo) * 2.0f;
    if (!(val >= 0.2f)) {
      px = __builtin_nanf("");
      py = __builtin_nanf("");
    }
    out[bid * 3 + 0] = px;
    out[bid * 3 + 1] = py;
    out[bid * 3 + 2] = val;
  }
}

extern "C" void kernel_launch(void* const* d_in, const int* in_sizes, int n_in,
                              void* d_out, int out_size, void* d_ws,
                              size_t ws_size, hipStream_t stream) {
  (void)in_sizes; (void)n_in; (void)out_size; (void)ws_size;
  const float* crops = (const float*)d_in[0];
  const float* W1 = (const float*)d_in[1];
  const float* b1 = (const float*)d_in[2];
  const float* W2 = (const float*)d_in[3];
  const float* b2 = (const float*)d_in[4];
  float* out = (float*)d_out;

  unsigned long long* keys = (unsigned long long*)d_ws;
  float* W2T = (float*)((char*)d_ws + (size_t)NKEYS * 8);  // 32x288 f32

  prep_kernel<<<36, 256, 0, stream>>>(W2, keys, W2T);
  convpeak_kernel<<<128 * 36, 128, 0, stream>>>(crops, W1, b1, b2, W2T, keys);
  refine_kernel<<<128 * 17, 32, 0, stream>>>(crops, W1, b1, W2, b2, keys, out);
}